// RenderingKaolin_15719580304402
// MI455X (gfx1250) — compile-verified
//
#include <hip/hip_runtime.h>
#include <hip/hip_bf16.h>
#include <math.h>

#define HH 128
#define WW 128
#define NFRAME 4
#define NV 256
#define NFACE 512
#define TEXN 256
#define CAMD 2.0f
#define SIGMAINV 7000.0f
#define PADR 3

typedef __attribute__((ext_vector_type(2))) float v2f;
typedef __attribute__((ext_vector_type(8))) float v8f;

#if defined(__HIP_DEVICE_COMPILE__) && !__has_builtin(__builtin_amdgcn_wmma_f32_16x16x4_f32)
#error "wmma_f32_16x16x4_f32 builtin not available in device pass"
#endif

// workspace layout (floats), per frame:
//   coef : 4 functionals (w0,w1,w2,z) x 512 faces x 4 floats (alpha,beta,gamma,0)
//   fdata: 512 x 8  (ax,ay,bx,by,cx,cy,fnz,areaValid)
//   fz   : 512 x 4  (z0,z1,z2,0)
//   soft : 128*128
#define COEF_OFF   0
#define FDATA_OFF  (4*NFACE*4)
#define FZ_OFF     (FDATA_OFF + NFACE*8)
#define SOFT_OFF   (FZ_OFF + NFACE*4)
#define FRAME_STRIDE (SOFT_OFF + HH*WW)

__global__ __launch_bounds__(256) void prep_kernel(
    const float* __restrict__ trans, const float* __restrict__ quat,
    const float* __restrict__ uverts, const int* __restrict__ faces,
    float* __restrict__ ws) {
  int frame = blockIdx.x;
  int tid = threadIdx.x;
  __shared__ float s_img[NV][2];
  __shared__ float s_cam[NV][3];

  float qw = quat[frame*4+0], qx = quat[frame*4+1], qy = quat[frame*4+2], qz = quat[frame*4+3];
  float qn = sqrtf(qw*qw + qx*qx + qy*qy + qz*qz);
  qw /= qn; qx /= qn; qy /= qn; qz /= qn;
  float R00 = 1.f-2.f*(qy*qy+qz*qz), R01 = 2.f*(qx*qy-qw*qz), R02 = 2.f*(qx*qz+qw*qy);
  float R10 = 2.f*(qx*qy+qw*qz), R11 = 1.f-2.f*(qx*qx+qz*qz), R12 = 2.f*(qy*qz-qw*qx);
  float R20 = 2.f*(qx*qz-qw*qy), R21 = 2.f*(qy*qz+qw*qx), R22 = 1.f-2.f*(qx*qx+qy*qy);
  float tx = trans[frame*3+0], ty = trans[frame*3+1], tz = trans[frame*3+2];
  float tanf_ = tanf(0.5f * (1.57f / 2.0f));   // tan(FOVY/2), ratio = W/H = 1

  // one vertex per thread (NV == blockDim)
  float ux = uverts[tid*3+0], uy = uverts[tid*3+1], uz = uverts[tid*3+2];
  float wx_ = R00*ux + R01*uy + R02*uz + tx;
  float wy_ = R10*ux + R11*uy + R12*uz + ty;
  float wz_ = R20*ux + R21*uy + R22*uz + tz;
  float pz = wz_ - CAMD;
  float negz = -pz;
  s_img[tid][0] = wx_ / tanf_ / negz;
  s_img[tid][1] = wy_ / tanf_ / negz;
  s_cam[tid][0] = wx_; s_cam[tid][1] = wy_; s_cam[tid][2] = pz;
  __syncthreads();

  float* wf = ws + (size_t)frame * FRAME_STRIDE;
  for (int f = tid; f < NFACE; f += blockDim.x) {
    int i0 = faces[f*3+0], i1 = faces[f*3+1], i2 = faces[f*3+2];
    float ax = s_img[i0][0], ay = s_img[i0][1];
    float bx = s_img[i1][0], by = s_img[i1][1];
    float cx = s_img[i2][0], cy = s_img[i2][1];
    float z0 = s_cam[i0][2], z1 = s_cam[i1][2], z2 = s_cam[i2][2];
    float e1x = s_cam[i1][0]-s_cam[i0][0], e1y = s_cam[i1][1]-s_cam[i0][1], e1z = z1-z0;
    float e2x = s_cam[i2][0]-s_cam[i0][0], e2y = s_cam[i2][1]-s_cam[i0][1], e2z = z2-z0;
    float nx = e1y*e2z - e1z*e2y, ny = e1z*e2x - e1x*e2z, nz = e1x*e2y - e1y*e2x;
    float fnz = nz / (sqrtf(nx*nx + ny*ny + nz*nz) + 1e-10f);
    float area = (bx-ax)*(cy-ay) - (by-ay)*(cx-ax);
    float areaValid = (fabsf(area) > 1e-10f) ? 1.0f : 0.0f;
    float as = (fabsf(area) < 1e-10f) ? 1e-10f : area;
    float inv = 1.0f / as;
    float a0 = (cy-by)*inv, b0 = -(cx-bx)*inv, g0 = (-bx*(cy-by) + by*(cx-bx))*inv;
    float a1 = (ay-cy)*inv, b1 = -(ax-cx)*inv, g1 = (-cx*(ay-cy) + cy*(ax-cx))*inv;
    float a2 = (by-ay)*inv, b2 = -(bx-ax)*inv, g2 = (-ax*(by-ay) + ay*(bx-ax))*inv;
    float az = a0*z0 + a1*z1 + a2*z2;
    float bz = b0*z0 + b1*z1 + b2*z2;
    float gz = g0*z0 + g1*z1 + g2*z2;
    float* c;
    c = wf + COEF_OFF + (0*NFACE + f)*4; c[0]=a0; c[1]=b0; c[2]=g0; c[3]=0.f;
    c = wf + COEF_OFF + (1*NFACE + f)*4; c[0]=a1; c[1]=b1; c[2]=g1; c[3]=0.f;
    c = wf + COEF_OFF + (2*NFACE + f)*4; c[0]=a2; c[1]=b2; c[2]=g2; c[3]=0.f;
    c = wf + COEF_OFF + (3*NFACE + f)*4; c[0]=az; c[1]=bz; c[2]=gz; c[3]=0.f;
    float* fd = wf + FDATA_OFF + f*8;
    fd[0]=ax; fd[1]=ay; fd[2]=bx; fd[3]=by; fd[4]=cx; fd[5]=cy; fd[6]=fnz; fd[7]=areaValid;
    float* fzp = wf + FZ_OFF + f*4;
    fzp[0]=z0; fzp[1]=z1; fzp[2]=z2; fzp[3]=0.f;
  }
}

__device__ __forceinline__ float segd2(float ax, float ay, float bx, float by,
                                       float px, float py) {
  float abx = bx-ax, aby = by-ay;
  float apx = px-ax, apy = py-ay;
  float t = (apx*abx + apy*aby) / (abx*abx + aby*aby + 1e-12f);
  t = fminf(fmaxf(t, 0.0f), 1.0f);
  float dx = apx - t*abx, dy = apy - t*aby;
  return dx*dx + dy*dy;
}

__global__ __launch_bounds__(32) void raster_kernel(
    const float* __restrict__ ffeat, const float* __restrict__ tex,
    float* __restrict__ out, float* __restrict__ ws) {
  int bid = blockIdx.x;            // NFRAME * 1024 tiles
  int frame = bid >> 10;
  int tile = bid & 1023;
  int y  = tile >> 3;              // pixel row
  int tX = tile & 7;               // 16-pixel column tile
  int lane = threadIdx.x;
  int half = lane >> 4;
  int ln = lane & 15;

  float* wf = ws + (size_t)frame * FRAME_STRIDE;
  const float stepx = 2.0f / WW;
  float py = 1.0f - (y + 0.5f) * (2.0f / HH);

  // A matrix (16 pixels x [px, py, 1, 0])
  v2f A;
  if (half == 0) { A.x = ((float)(tX*16 + ln) + 0.5f)*stepx - 1.0f; A.y = py; }
  else           { A.x = 1.0f;                                      A.y = 0.0f; }

  float bestZ[8], softS[8];
  int bestF[8];
#pragma unroll
  for (int k = 0; k < 8; ++k) { bestZ[k] = -1e30f; softS[k] = 0.0f; bestF[k] = -1; }

  float pxBase = ((float)(tX*16 + half*8) + 0.5f)*stepx - 1.0f;
  const float* cbase = wf + COEF_OFF;
  int co = half * 2;   // lanes 0-15 read (alpha,beta); 16-31 read (gamma,0)

  for (int ft = 0; ft < NFACE/16; ++ft) {
    int face = ft*16 + ln;
    v2f B0 = { cbase[(0*NFACE+face)*4 + co], cbase[(0*NFACE+face)*4 + co + 1] };
    v2f B1 = { cbase[(1*NFACE+face)*4 + co], cbase[(1*NFACE+face)*4 + co + 1] };
    v2f B2 = { cbase[(2*NFACE+face)*4 + co], cbase[(2*NFACE+face)*4 + co + 1] };
    v2f Bz = { cbase[(3*NFACE+face)*4 + co], cbase[(3*NFACE+face)*4 + co + 1] };
    v8f zero = {0.f,0.f,0.f,0.f,0.f,0.f,0.f,0.f};
#if defined(__HIP_DEVICE_COMPILE__)
    v8f dw0 = __builtin_amdgcn_wmma_f32_16x16x4_f32(false, A, false, B0, (short)0, zero, false, false);
    v8f dw1 = __builtin_amdgcn_wmma_f32_16x16x4_f32(false, A, false, B1, (short)0, zero, false, false);
    v8f dw2 = __builtin_amdgcn_wmma_f32_16x16x4_f32(false, A, false, B2, (short)0, zero, false, false);
    v8f dz  = __builtin_amdgcn_wmma_f32_16x16x4_f32(false, A, false, Bz, (short)0, zero, false, false);
#else
    v8f dw0 = zero, dw1 = zero, dw2 = zero, dz = zero;
#endif

    const float* fd = wf + FDATA_OFF + face*8;
    float ax = fd[0], ay = fd[1], bx = fd[2], by = fd[3], cx = fd[4], cy = fd[5];
    float fnz = fd[6], av = fd[7];
    float fpos = (fnz > 0.0f) ? 1.0f : 0.0f;

#pragma unroll
    for (int k = 0; k < 8; ++k) {
      float w0 = dw0[k], w1 = dw1[k], w2 = dw2[k], z = dz[k];
      bool inside = (w0 >= 0.f) && (w1 >= 0.f) && (w2 >= 0.f) && (av > 0.5f);
      bool take = inside && (fpos > 0.5f) && (z > bestZ[k]);
      bestZ[k] = take ? z    : bestZ[k];
      bestF[k] = take ? face : bestF[k];
      float px = pxBase + (float)k * stepx;
      float d2 = fminf(fminf(segd2(ax,ay,bx,by,px,py), segd2(bx,by,cx,cy,px,py)),
                       segd2(cx,cy,ax,ay,px,py));
      d2 = inside ? 0.0f : d2;
      float prob = fminf(expf(-d2 * SIGMAINV), 1.0f - 1e-4f) * fpos;
      softS[k] += log1pf(-prob);
    }
  }

  // reduce across the 16-lane face group: argmax(z) + sum(softS)
#pragma unroll
  for (int off = 1; off < 16; off <<= 1) {
#pragma unroll
    for (int k = 0; k < 8; ++k) {
      float oz = __shfl_xor(bestZ[k], off, 16);
      int   of = __shfl_xor(bestF[k], off, 16);
      float os = __shfl_xor(softS[k], off, 16);
      softS[k] += os;
      bool take = (oz > bestZ[k]) ||
                  (oz == bestZ[k] && (unsigned)of < (unsigned)bestF[k]);
      bestZ[k] = take ? oz : bestZ[k];
      bestF[k] = take ? of : bestF[k];
    }
  }

  // finalize: lane ln==k handles pixel (tX*16 + half*8 + k)
#pragma unroll
  for (int k = 0; k < 8; ++k) {
    if (ln == k) {
      int x = tX*16 + half*8 + k;
      float px = ((float)x + 0.5f)*stepx - 1.0f;
      int best = bestF[k];
      bool has = best >= 0;
      int redw = has ? best : (NFACE - 1);
      float u = 0.f, v = 0.f;
      if (has) {
        const float* c0 = wf + COEF_OFF + (0*NFACE + best)*4;
        const float* c1 = wf + COEF_OFF + (1*NFACE + best)*4;
        const float* c2 = wf + COEF_OFF + (2*NFACE + best)*4;
        float w0 = c0[0]*px + c0[1]*py + c0[2];
        float w1 = c1[0]*px + c1[1]*py + c1[2];
        float w2 = c2[0]*px + c2[1]*py + c2[2];
        const float* ff = ffeat + best*6;
        u = w0*ff[0] + w1*ff[2] + w2*ff[4];
        v = w0*ff[1] + w1*ff[3] + w2*ff[5];
      }
      float uu = fminf(fmaxf(u, 0.f), 1.f);
      float vv = fminf(fmaxf(v, 0.f), 1.f);
      float xf = uu*(float)TEXN - 0.5f;
      float yf = (1.0f - vv)*(float)TEXN - 0.5f;
      float x0 = floorf(xf), y0 = floorf(yf);
      float wx = xf - x0, wy = yf - y0;
      int x0i = min(max((int)x0, 0), TEXN-1);
      int x1i = min(x0i + 1, TEXN-1);
      int y0i = min(max((int)y0, 0), TEXN-1);
      int y1i = min(y0i + 1, TEXN-1);
#pragma unroll
      for (int ch = 0; ch < 3; ++ch) {
        const float* tc = tex + ch*TEXN*TEXN;
        float t00 = tc[y0i*TEXN + x0i], t01 = tc[y0i*TEXN + x1i];
        float t10 = tc[y1i*TEXN + x0i], t11 = tc[y1i*TEXN + x1i];
        float rgb = (1.f-wy)*((1.f-wx)*t00 + wx*t01) + wy*((1.f-wx)*t10 + wx*t11);
        out[(((size_t)frame*5 + ch)*HH + y)*WW + x] = rgb;
      }
      const float* fzp = wf + FZ_OFF + redw*4;
      float depth = (fzp[0] + fzp[1] + fzp[2]) * (1.0f/3.0f);
      out[(((size_t)frame*5 + 4)*HH + y)*WW + x] = depth;
      float soft = 1.0f - expf(softS[k]);
      wf[SOFT_OFF + y*WW + x] = soft;
    }
  }
}

__global__ __launch_bounds__(256) void erode_kernel(const float* __restrict__ ws,
                                                    float* __restrict__ out) {
  int idx = blockIdx.x * blockDim.x + threadIdx.x;
  if (idx >= NFRAME*HH*WW) return;
  int frame = idx / (HH*WW);
  int p = idx % (HH*WW);
  int y = p / WW, x = p % WW;
  const float* soft = ws + (size_t)frame*FRAME_STRIDE + SOFT_OFF;
  float m = __builtin_inff();
  for (int dy = -PADR; dy <= PADR; ++dy) {
    int yy = y + dy;
    if (yy < 0 || yy >= HH) continue;
    for (int dx = -PADR; dx <= PADR; ++dx) {
      int xx = x + dx;
      if (xx < 0 || xx >= WW) continue;
      m = fminf(m, soft[yy*WW + xx]);
    }
  }
  out[(((size_t)frame*5 + 3)*HH + y)*WW + x] = m;
}

extern "C" void kernel_launch(void* const* d_in, const int* in_sizes, int n_in,
                              void* d_out, int out_size, void* d_ws, size_t ws_size,
                              hipStream_t stream) {
  const float* trans  = (const float*)d_in[0];   // (1,1,4,3)
  const float* quat   = (const float*)d_in[1];   // (1,4,4)
  const float* uverts = (const float*)d_in[2];   // (1,256,3)
  const float* ffeat  = (const float*)d_in[3];   // (1,512,3,2)
  const float* tex    = (const float*)d_in[4];   // (1,3,256,256)
  const int*   faces  = (const int*)d_in[5];     // (512,3)
  float* out = (float*)d_out;                    // (1,4,1,5,128,128)
  float* ws  = (float*)d_ws;

  prep_kernel<<<NFRAME, 256, 0, stream>>>(trans, quat, uverts, faces, ws);
  raster_kernel<<<NFRAME * 1024, 32, 0, stream>>>(ffeat, tex, out, ws);
  erode_kernel<<<(NFRAME*HH*WW + 255) / 256, 256, 0, stream>>>(ws, out);
}